// TrueQuantizationLayer_10153302688317
// MI455X (gfx1250) — compile-verified
//
#include <hip/hip_runtime.h>

// SAR-ADC 4-bit quantizer, MI455X (gfx1250, wave32).
// Pure streaming elementwise kernel: ~704 MB traffic, ~30 us at 23.3 TB/s HBM.
// No matrix structure -> no WMMA (would only add data movement).
// Data path: NT b32 loads of x, NT b32 stores of q, NT b128 stores of the 4
// Q bits per element; weights via the uniform scalar path (s_load / K$).

typedef __attribute__((ext_vector_type(4))) float v4f;

static __device__ __forceinline__ float signf(float v) {
    // jnp.sign semantics for finite inputs: -1 / 0 / +1
    return (v > 0.0f) ? 1.0f : ((v < 0.0f) ? -1.0f : 0.0f);
}

__global__ __launch_bounds__(256) void sar_adc_quant_kernel(
    const float* __restrict__ x,    // [N] flattened [L, 28]
    const float* __restrict__ W,    // [10]
    float*       __restrict__ q_out, // [N]
    v4f*         __restrict__ Q_out, // [N] x 4 floats (bit index fastest)
    int n)
{
    const float VRc = 0.0625f;   // VDD / 2^Q_BITS
    const float DLT = 1e-30f;    // reference DELTA

    const int e = blockIdx.x * blockDim.x + threadIdx.x;
    if (e >= n) return;

    // Uniform weight loads -> scalar constant-cache path.
    const float w0 = W[0], w1 = W[1], w2 = W[2], w3 = W[3], w4 = W[4];
    const float w5 = W[5], w6 = W[6], w7 = W[7], w8 = W[8], w9 = W[9];

    const float xv = __builtin_nontemporal_load(&x[e]);

    // Replicate reference FP evaluation order exactly; __f*_rn intrinsics
    // block -ffp-contract=fast so no FMAs alter the sign() boundaries.

    // j = 3: bit_sum = W[9]*VREF
    float bs = __fmul_rn(w9, VRc);
    const float Q3 = signf(__fadd_rn(__fsub_rn(xv, bs), DLT));
    const float p3 = __fmul_rn(__fadd_rn(Q3, 1.0f), 0.5f);   // exact {0,.5,1}

    // j = 2: bit_sum = W[8]*VREF + p3*W[7]*VR
    bs = __fmul_rn(w8, VRc);
    bs = __fadd_rn(bs, __fmul_rn(__fmul_rn(p3, w7), VRc));
    const float Q2 = signf(__fadd_rn(__fsub_rn(xv, bs), DLT));
    const float p2 = __fmul_rn(__fadd_rn(Q2, 1.0f), 0.5f);

    // j = 1: bit_sum = W[6]*VREF + p2*W[5]*VR + p3*W[4]*VR
    bs = __fmul_rn(w6, VRc);
    bs = __fadd_rn(bs, __fmul_rn(__fmul_rn(p2, w5), VRc));
    bs = __fadd_rn(bs, __fmul_rn(__fmul_rn(p3, w4), VRc));
    const float Q1 = signf(__fadd_rn(__fsub_rn(xv, bs), DLT));
    const float p1 = __fmul_rn(__fadd_rn(Q1, 1.0f), 0.5f);

    // j = 0: bit_sum = W[3]*VREF + p1*W[2]*VR + p2*W[1]*VR + p3*W[0]*VR
    bs = __fmul_rn(w3, VRc);
    bs = __fadd_rn(bs, __fmul_rn(__fmul_rn(p1, w2), VRc));
    bs = __fadd_rn(bs, __fmul_rn(__fmul_rn(p2, w1), VRc));
    bs = __fadd_rn(bs, __fmul_rn(__fmul_rn(p3, w0), VRc));
    const float Q0 = signf(__fadd_rn(__fsub_rn(xv, bs), DLT));
    const float p0 = __fmul_rn(__fadd_rn(Q0, 1.0f), 0.5f);

    // q = sum_b p_b * VR * 2^b  -- every term is a multiple of 2^-5, so this
    // is exact in fp32 regardless of accumulation order.
    const float q = (p0 + 2.0f * p1 + 4.0f * p2 + 8.0f * p3) * VRc;

    // Streaming NT stores (output working set >> 192 MB L2).
    __builtin_nontemporal_store(q, &q_out[e]);
    v4f Qv = { Q0, Q1, Q2, Q3 };
    __builtin_nontemporal_store(Qv, &Q_out[e]);
}

extern "C" void kernel_launch(void* const* d_in, const int* in_sizes, int n_in,
                              void* d_out, int out_size, void* d_ws, size_t ws_size,
                              hipStream_t stream) {
    const float* x = (const float*)d_in[0];   // [LENGTH * NUM_ADCS] fp32
    const float* W = (const float*)d_in[1];   // [10] fp32
    const int n = in_sizes[0];                // LENGTH * NUM_ADCS = 29,360,128

    // d_out layout (tuple return, concatenated flat):
    //   [0, n)        : q  [L, 28]
    //   [n, 5n)       : Q  [L, 28, 4]
    float* q_out = (float*)d_out;
    v4f*   Q_out = (v4f*)((float*)d_out + (size_t)n);  // n*4 bytes offset, 16B aligned

    const int threads = 256;                  // 8 wave32 waves per block
    const int blocks = (n + threads - 1) / threads;
    sar_adc_quant_kernel<<<blocks, threads, 0, stream>>>(x, W, q_out, Q_out, n);
}